// Noise_3384434229807
// MI455X (gfx1250) — compile-verified
//
#include <hip/hip_runtime.h>
#include <stdint.h>

#define THREADS 256
#define PTS_PER_BLOCK 1024
#define TILE_FLOATS (PTS_PER_BLOCK * 3)         // 3072 floats per array
#define TILE_BYTES  (TILE_FLOATS * 4)           // 12288 bytes per array
#define XFER_ITERS  (TILE_BYTES / 16 / THREADS) // 3 x b128 per thread per array

__device__ __forceinline__ float hash_grad(float t_over_2pi) {
    // sin(2*pi*fract(x)) == sin(2*pi*x); keeps v_sin_f32 in-range.
    float f = __builtin_amdgcn_fractf(t_over_2pi);
    float s = __sinf(f * 6.2831853071795864f);
    float h = s * 43758.5453f;
    float fr = __builtin_amdgcn_fractf(h);      // h - floor(h), matches jnp
    return fmaf(fr, 2.0f, -1.0f);               // [-1, 1)
}

__device__ __forceinline__ float point_noise(const float* __restrict__ smCell,
                                             const float* __restrict__ smFrac,
                                             int pl, float sd) {
    const float INV2PI = 0.15915494309189535f;
    const float K00 = 127.1f, K01 = 311.7f, K02 = 74.7f;
    const float K10 = 269.5f, K11 = 183.3f, K12 = 246.1f;
    const float K20 = 113.5f, K21 = 271.9f, K22 = 124.6f;

    const float cx = smCell[3 * pl + 0];
    const float cy = smCell[3 * pl + 1];
    const float cz = smCell[3 * pl + 2];
    const float fx = smFrac[3 * pl + 0];
    const float fy = smFrac[3 * pl + 1];
    const float fz = smFrac[3 * pl + 2];

    // base_g = (K_g . cell + seed) / (2*pi)   (all the "matmul" that survives)
    const float b0 = (fmaf(cx, K00, fmaf(cy, K01, cz * K02)) + sd) * INV2PI;
    const float b1 = (fmaf(cx, K10, fmaf(cy, K11, cz * K12)) + sd) * INV2PI;
    const float b2 = (fmaf(cx, K20, fmaf(cy, K21, cz * K22)) + sd) * INV2PI;

    const float wx[2] = {1.0f - fx, fx};
    const float wy[2] = {1.0f - fy, fy};
    const float wz[2] = {1.0f - fz, fz};
    const float dx[2] = {fx, fx - 1.0f};
    const float dy[2] = {fy, fy - 1.0f};
    const float dz[2] = {fz, fz - 1.0f};

    float res = 0.0f;
#pragma unroll
    for (int c = 0; c < 8; ++c) {
        const int bx = (c >> 2) & 1, by = (c >> 1) & 1, bz = c & 1;
        // K_g . corner -> compile-time constants (pre-divided by 2*pi)
        const float o0 = ((float)bx * K00 + (float)by * K01 + (float)bz * K02) * INV2PI;
        const float o1 = ((float)bx * K10 + (float)by * K11 + (float)bz * K12) * INV2PI;
        const float o2 = ((float)bx * K20 + (float)by * K21 + (float)bz * K22) * INV2PI;

        const float g0 = hash_grad(b0 + o0);
        const float g1 = hash_grad(b1 + o1);
        const float g2 = hash_grad(b2 + o2);

        const float dot = fmaf(g0, dx[bx], fmaf(g1, dy[by], g2 * dz[bz]));
        const float w = wx[bx] * wy[by] * wz[bz];
        res = fmaf(w, dot, res);
    }
    return res;
}

__global__ __launch_bounds__(THREADS) void Noise_kernel(
    const float* __restrict__ cell,   // [P*3]
    const float* __restrict__ frac,   // [P*3]
    const float* __restrict__ seedp,  // [B]
    float* __restrict__ out,          // [P]
    int P, int N)                     // N = points per texture
{
    __shared__ __align__(16) float smCell[TILE_FLOATS];
    __shared__ __align__(16) float smFrac[TILE_FLOATS];

    const int t = threadIdx.x;
    const long long tileStart  = (long long)blockIdx.x * PTS_PER_BLOCK;
    const long long tileByte   = tileStart * 12;
    const long long arrayBytes = (long long)P * 12;

    const unsigned ldsC = (unsigned)(uintptr_t)(&smCell[0]) + (unsigned)(t * 16);
    const unsigned ldsF = (unsigned)(uintptr_t)(&smFrac[0]) + (unsigned)(t * 16);

    // ---- Async global -> LDS staging (ASYNCcnt path, b128/lane) ----
    // IOFFSET applies to BOTH the LDS and global address (CDNA5 ISA
    // 4.4), so the 3 strided transfers per array share one address VGPR.
    if (tileByte + TILE_BYTES <= arrayBytes) {   // uniform: full tile, no guards
        const int goff = (int)tileByte + t * 16; // P*12 < 2^31
        asm volatile("global_load_async_to_lds_b128 %0, %1, %2"
                     :: "v"(ldsC), "v"(goff), "s"(cell) : "memory");
        asm volatile("global_load_async_to_lds_b128 %0, %1, %2 offset:4096"
                     :: "v"(ldsC), "v"(goff), "s"(cell) : "memory");
        asm volatile("global_load_async_to_lds_b128 %0, %1, %2 offset:8192"
                     :: "v"(ldsC), "v"(goff), "s"(cell) : "memory");
        asm volatile("global_load_async_to_lds_b128 %0, %1, %2"
                     :: "v"(ldsF), "v"(goff), "s"(frac) : "memory");
        asm volatile("global_load_async_to_lds_b128 %0, %1, %2 offset:4096"
                     :: "v"(ldsF), "v"(goff), "s"(frac) : "memory");
        asm volatile("global_load_async_to_lds_b128 %0, %1, %2 offset:8192"
                     :: "v"(ldsF), "v"(goff), "s"(frac) : "memory");
    } else {                                     // tail tile: per-lane guards
#pragma unroll
        for (int i = 0; i < XFER_ITERS; ++i) {
            long long byteInTile = (long long)(i * THREADS + t) * 16;
            long long gbyte = tileByte + byteInTile;
            if (gbyte + 16 <= arrayBytes) {
                unsigned lc = ldsC + (unsigned)(i * 4096);
                unsigned lf = ldsF + (unsigned)(i * 4096);
                int goff = (int)gbyte;
                asm volatile("global_load_async_to_lds_b128 %0, %1, %2"
                             :: "v"(lc), "v"(goff), "s"(cell) : "memory");
                asm volatile("global_load_async_to_lds_b128 %0, %1, %2"
                             :: "v"(lf), "v"(goff), "s"(frac) : "memory");
            }
        }
    }
    asm volatile("s_wait_asynccnt 0x0" ::: "memory");
    __syncthreads();

    // ---- Per-block uniform seed (PTS_PER_BLOCK divides N) ----
    const int b = (int)(tileStart / (long long)N);
    const float sd = seedp[b];

    if (tileStart + PTS_PER_BLOCK <= (long long)P) {   // uniform fast path
#pragma unroll
        for (int j = 0; j < PTS_PER_BLOCK / THREADS; ++j) {
            const int pl = t + j * THREADS;            // bank stride 3: no conflicts
            out[tileStart + pl] = point_noise(smCell, smFrac, pl, sd);
        }
    } else {
#pragma unroll
        for (int j = 0; j < PTS_PER_BLOCK / THREADS; ++j) {
            const int pl = t + j * THREADS;
            const long long p = tileStart + pl;
            if (p < (long long)P)
                out[p] = point_noise(smCell, smFrac, pl, sd);
        }
    }
}

extern "C" void kernel_launch(void* const* d_in, const int* in_sizes, int n_in,
                              void* d_out, int out_size, void* d_ws, size_t ws_size,
                              hipStream_t stream) {
    (void)n_in; (void)d_ws; (void)ws_size; (void)out_size;
    const float* pos  = (const float*)d_in[0];   // [2, B, N, 3]
    const float* seed = (const float*)d_in[1];   // [B]
    const int P = in_sizes[0] / 6;               // B*N points
    const int B = in_sizes[1];
    const int N = P / B;
    const float* cell = pos;                     // position[0]
    const float* frac = pos + (size_t)P * 3;     // position[1]
    float* out = (float*)d_out;                  // [B, N]

    const int blocks = (P + PTS_PER_BLOCK - 1) / PTS_PER_BLOCK;
    Noise_kernel<<<blocks, THREADS, 0, stream>>>(cell, frac, seed, out, P, N);
}